// YOLOLoss_10685878633086
// MI455X (gfx1250) — compile-verified
//
#include <hip/hip_runtime.h>

#define W_ 52
#define H_ 52
#define A_ 3
#define B_ 16
#define NCH 85            // 5 + 80 classes
#define HW 2704           // 52*52
#define NT 64             // targets
#define TILE 128          // cells per block
#define NBLK 1014         // 16*3*2704 / 128
#define CS 132            // padded LDS column stride (multiple of 4 -> 16B-aligned rows)
#define NOUT 11032320     // 16 * 3*2704 * 85
#define STRIDE8 8.0f      // 416/52

__device__ __forceinline__ float sigf(float v) { return 1.0f / (1.0f + __expf(-v)); }

// ---------------------------------------------------------------------------
// Kernel 1: per-target precompute + per-image target ranges
// wsi layout (ints):   [0,64) img_s  [64,128) best  [128,192) gj  [192,256) gi
//                      [256,320) label  [320,336) tstart  [336,352) tend
// wsf layout (floats): [0,64) tx  [64,128) ty  [128,192) tw  [192,256) th  [256,320) sc
// ---------------------------------------------------------------------------
__global__ void yolo_setup(const float* __restrict__ target,
                           const float* __restrict__ anchors,
                           int* __restrict__ wsi, float* __restrict__ wsf) {
  const int t = threadIdx.x;
  if (t < NT) {
    const float timg = target[t * 6 + 0];
    const float txn = target[t * 6 + 1], tyn = target[t * 6 + 2];
    const float twn = target[t * 6 + 3], thn = target[t * 6 + 4];
    const int label = (int)target[t * 6 + 5];
    const float gx = txn * W_, gy = tyn * H_, gw = twn * W_, gh = thn * H_;
    const int gi = (int)gx, gj = (int)gy;
    int best = 0; float bv = -1.0f;
    #pragma unroll
    for (int k = 0; k < A_; ++k) {
      const float aw = anchors[2 * k] * (1.0f / 416.0f);
      const float ah = anchors[2 * k + 1] * (1.0f / 416.0f);
      const float inter = fminf(aw, twn) * fminf(ah, thn);
      const float iou = inter / (aw * ah + twn * thn - inter + 1e-16f);
      if (iou > bv) { bv = iou; best = k; }
    }
    const int img = (int)timg;
    const bool valid = (gi >= 0) && (gi < W_) && (gj >= 0) && (gj < H_);
    wsi[t]          = valid ? img : B_;
    wsi[NT + t]     = best;
    wsi[2 * NT + t] = gj;
    wsi[3 * NT + t] = gi;
    wsi[4 * NT + t] = label;
    wsf[t]          = gx - floorf(gx);
    wsf[NT + t]     = gy - floorf(gy);
    wsf[2 * NT + t] = __logf(gw * STRIDE8 / anchors[2 * best] + 1e-16f);
    wsf[3 * NT + t] = __logf(gh * STRIDE8 / anchors[2 * best + 1] + 1e-16f);
    wsf[4 * NT + t] = 2.0f - twn * thn;
  } else if (t < NT + B_) {
    const int b = t - NT;
    int st = NT, en = 0;
    for (int n = 0; n < NT; ++n) {
      if ((int)target[n * 6] == b) { if (n < st) st = n; en = n + 1; }
    }
    if (en == 0) st = 0;
    wsi[5 * NT + b]      = st;
    wsi[5 * NT + B_ + b] = en;
  }
}

// ---------------------------------------------------------------------------
// Kernel 2: streaming transform + transpose + per-block loss partials.
// One block = 128 consecutive cells. Channel-major reads are staged into LDS
// with CDNA5 async B128 global->LDS copies (one instruction = one 512B channel
// row); the transposed output tile leaves via async LDS->global stores.
// ---------------------------------------------------------------------------
__global__ __launch_bounds__(TILE) void yolo_main(
    const float* __restrict__ x,
    const float* __restrict__ target,
    const float* __restrict__ anchors,
    const int* __restrict__ wsi,
    const float* __restrict__ wsf,
    float* __restrict__ partials,
    float* __restrict__ out) {
  __shared__ float lds[NCH * CS];
  __shared__ float red[4 * (TILE / 32)];

  const int t = threadIdx.x;
  const int lane = t & 31;
  const int wv = t >> 5;
  const int gc = blockIdx.x * TILE + t;
  const int ba = gc / HW;
  const int s  = gc - ba * HW;
  const int b  = ba / A_;
  const int a  = ba - b * A_;
  const int jj = s / W_;
  const int ii = s - jj * W_;
  const unsigned ldsBase = (unsigned)(unsigned long long)&lds[0];

  // ---- async B128 copy: wave wv loads channels c = wv, wv+4, ... ----------
  // Lane covers cells [4*lane, 4*lane+4) of the tile (never straddles a
  // (b,a) boundary since 2704 % 4 == 0).
  {
    const int cell0 = blockIdx.x * TILE + 4 * lane;
    const int ba_l = cell0 / HW;
    const int s_l  = cell0 - ba_l * HW;
    const unsigned vbase = (unsigned)((ba_l * NCH * HW + s_l) * 4);
    for (int c = wv; c < NCH; c += 4) {
      const unsigned vo = vbase + (unsigned)(c * (HW * 4));
      const unsigned la = ldsBase + (unsigned)((c * CS + 4 * lane) * 4);
      asm volatile("global_load_async_to_lds_b128 %0, %1, %2"
                   :: "v"(la), "v"(vo), "s"(x)
                   : "memory");
    }
  }
  asm volatile("s_wait_asynccnt 0" ::: "memory");
  __syncthreads();   // other waves loaded 3 of every 4 channels of our column

  // ---- in-place transform of own column -----------------------------------
  const float r0 = lds[t];
  const float r1 = lds[CS + t];
  const float r2 = lds[2 * CS + t];
  const float r3 = lds[3 * CS + t];
  const float r4 = lds[4 * CS + t];
  const float aw = anchors[2 * a], ah = anchors[2 * a + 1];
  const float px = sigf(r0), py = sigf(r1), pconf = sigf(r4);
  const float bx = px + (float)ii, by = py + (float)jj;
  const float bw = __expf(r2) * (aw * 0.125f);   // exp(pw)*scaled_anchor
  const float bh = __expf(r3) * (ah * 0.125f);
  lds[t]          = bx * STRIDE8;
  lds[CS + t]     = by * STRIDE8;
  lds[2 * CS + t] = bw * STRIDE8;
  lds[3 * CS + t] = bh * STRIDE8;
  lds[4 * CS + t] = pconf;
  float clsq = 0.0f;
  #pragma unroll 5
  for (int c = 5; c < NCH; ++c) {
    const float v = sigf(lds[c * CS + t]);
    lds[c * CS + t] = v;
    clsq += v * v;
  }

  // ---- per-cell loss contributions ----------------------------------------
  const float inv = 1.0f / (float)W_;
  const float nx = bx * inv, ny = by * inv, nw = bw * inv, nh = bh * inv;
  const float b1x1 = nx - 0.5f * nw, b1x2 = nx + 0.5f * nw;
  const float b1y1 = ny - 0.5f * nh, b1y2 = ny + 0.5f * nh;
  const int ts = wsi[5 * NT + b], te = wsi[5 * NT + B_ + b];
  int hit = -1;
  bool over = false;                 // any IoU > 0.7 (division-free)
  for (int n = ts; n < te; ++n) {
    const float tbx = target[n * 6 + 1], tby = target[n * 6 + 2];
    const float tbw = target[n * 6 + 3], tbh = target[n * 6 + 4];
    const float b2x1 = tbx - 0.5f * tbw, b2x2 = tbx + 0.5f * tbw;
    const float b2y1 = tby - 0.5f * tbh, b2y2 = tby + 0.5f * tbh;
    const float iw = fmaxf(fminf(b1x2, b2x2) - fmaxf(b1x1, b2x1), 0.0f);
    const float ih = fmaxf(fminf(b1y2, b2y2) - fmaxf(b1y1, b2y1), 0.0f);
    const float inter = iw * ih;
    const float uni = nw * nh + tbw * tbh - inter + 1e-16f;
    over = over || (inter > 0.7f * uni);
    if (wsi[n] == b && wsi[NT + n] == a && wsi[2 * NT + n] == jj &&
        wsi[3 * NT + n] == ii) hit = n;   // last match wins (scatter semantics)
  }
  float lxy = 0.0f, lwh = 0.0f, lconf = 0.0f, lcls = 0.0f;
  if (hit >= 0) {
    const float scv = wsf[4 * NT + hit];
    const float sc2 = scv * scv;
    const float dx = px - wsf[hit],          dy = py - wsf[NT + hit];
    const float dw = r2 - wsf[2 * NT + hit], dh = r3 - wsf[3 * NT + hit];
    lxy = (dx * dx + dy * dy) * sc2;
    lwh = (dw * dw + dh * dh) * sc2;
    const float dc = pconf - 1.0f;
    lconf = dc * dc;
    const int L = wsi[4 * NT + hit];
    const float pL = lds[(5 + L) * CS + t];
    lcls = clsq - 2.0f * pL + 1.0f;          // one-hot class target
  } else if (!over) {
    lconf = pconf * pconf;                   // noobj term, tconf = 0
  }

  __syncthreads();

  // ---- transposed tile writeout via async LDS->global stores --------------
  // out element e = cell*85 + c within the tile; output region is contiguous,
  // the transpose gather happens inside the async/LDS path (no VGPR staging).
  {
    const unsigned gBase = (unsigned)((size_t)blockIdx.x * (TILE * NCH) * 4);
    int e = t, c = t % NCH, cell = t / NCH;
    #pragma unroll 5
    for (int k = 0; k < NCH; ++k) {
      const unsigned lo = ldsBase + (unsigned)((c * CS + cell) * 4);
      const unsigned go = gBase + (unsigned)(e * 4);
      asm volatile("global_store_async_from_lds_b32 %0, %1, %2"
                   :: "v"(go), "v"(lo), "s"(out)
                   : "memory");
      e += TILE; c += TILE - NCH; cell += 1;  // 128 = 85 + 43
      if (c >= NCH) { c -= NCH; cell += 1; }
    }
  }
  // Outstanding async stores are flushed by S_ENDPGM's implicit wait-idle.

  // ---- deterministic block reduction of the four loss components ----------
  #pragma unroll
  for (int off = 16; off > 0; off >>= 1) {
    lxy   += __shfl_down(lxy,   off, 32);
    lwh   += __shfl_down(lwh,   off, 32);
    lconf += __shfl_down(lconf, off, 32);
    lcls  += __shfl_down(lcls,  off, 32);
  }
  const int ln = t & 31;
  if (ln == 0) {
    red[wv] = lxy; red[4 + wv] = lwh; red[8 + wv] = lconf; red[12 + wv] = lcls;
  }
  __syncthreads();
  if (t == 0) {
    float sxy = 0, swh = 0, scf = 0, scl = 0;
    #pragma unroll
    for (int w2 = 0; w2 < TILE / 32; ++w2) {
      sxy += red[w2]; swh += red[4 + w2]; scf += red[8 + w2]; scl += red[12 + w2];
    }
    partials[blockIdx.x * 4 + 0] = sxy;
    partials[blockIdx.x * 4 + 1] = swh;
    partials[blockIdx.x * 4 + 2] = scf;
    partials[blockIdx.x * 4 + 3] = scl;
  }
}

// ---------------------------------------------------------------------------
// Kernel 3: fixed-order final reduction -> 5 scalar losses
// ---------------------------------------------------------------------------
__global__ __launch_bounds__(256) void yolo_reduce(const float* __restrict__ partials,
                                                   float* __restrict__ out) {
  __shared__ float red[4 * 8];
  const int t = threadIdx.x;
  float s0 = 0, s1 = 0, s2 = 0, s3 = 0;
  for (int n = t; n < NBLK; n += 256) {
    s0 += partials[n * 4 + 0]; s1 += partials[n * 4 + 1];
    s2 += partials[n * 4 + 2]; s3 += partials[n * 4 + 3];
  }
  #pragma unroll
  for (int off = 16; off > 0; off >>= 1) {
    s0 += __shfl_down(s0, off, 32); s1 += __shfl_down(s1, off, 32);
    s2 += __shfl_down(s2, off, 32); s3 += __shfl_down(s3, off, 32);
  }
  const int wv = t >> 5, ln = t & 31;
  if (ln == 0) { red[wv] = s0; red[8 + wv] = s1; red[16 + wv] = s2; red[24 + wv] = s3; }
  __syncthreads();
  if (t == 0) {
    float a0 = 0, a1 = 0, a2 = 0, a3 = 0;
    #pragma unroll
    for (int w2 = 0; w2 < 8; ++w2) {
      a0 += red[w2]; a1 += red[8 + w2]; a2 += red[16 + w2]; a3 += red[24 + w2];
    }
    out[NOUT + 0] = a0 + a1 + a2 + a3;  // total
    out[NOUT + 1] = a0;                 // loss_xy
    out[NOUT + 2] = a1;                 // loss_wh
    out[NOUT + 3] = a2;                 // loss_conf
    out[NOUT + 4] = a3;                 // loss_cls
  }
}

extern "C" void kernel_launch(void* const* d_in, const int* in_sizes, int n_in,
                              void* d_out, int out_size, void* d_ws, size_t ws_size,
                              hipStream_t stream) {
  const float* x       = (const float*)d_in[0];
  const float* target  = (const float*)d_in[1];
  const float* anchors = (const float*)d_in[2];
  float* out = (float*)d_out;

  int*   wsi      = (int*)d_ws;                                // 352 ints
  float* wsf      = (float*)((char*)d_ws + 352 * sizeof(int)); // 320 floats
  float* partials = wsf + 5 * NT;                              // NBLK*4 floats

  yolo_setup<<<1, 128, 0, stream>>>(target, anchors, wsi, wsf);
  yolo_main<<<NBLK, TILE, 0, stream>>>(x, target, anchors, wsi, wsf, partials, out);
  yolo_reduce<<<1, 256, 0, stream>>>(partials, out);
}